// MobileNetV2_Ft_LinearFixed_30726196036197
// MI455X (gfx1250) — compile-verified
//
#include <hip/hip_runtime.h>
#include <hip/hip_bf16.h>

typedef __attribute__((ext_vector_type(2))) float v2f;
typedef __attribute__((ext_vector_type(8))) float v8f;

#define EPS_F 1e-7f

#define NB   8
#define DD   64
#define HH   256
#define WW   256
#define CC   21
#define HW   (HH * WW)        /* 65536 */
#define NPTS (NB * HW)        /* 524288 points */
#define TILE 128              /* points per block (half a w-row) */
#define SXS  66               /* padded LDS row stride (floats) */

// D = A(16x4 f32) * B(4x16 f32) + C(16x16 f32)
static __device__ __forceinline__ v8f wmma_f32x4(v2f a, v2f b, v8f c) {
  return __builtin_amdgcn_wmma_f32_16x16x4_f32(
      /*neg_a=*/false, a, /*neg_b=*/false, b,
      /*c_mod=*/(short)0, c, /*reuse_a=*/false, /*reuse_b=*/false);
}

// ---------------------------------------------------------------------------
// Prep: build B-matrix fragment stream (augmented K'=128, classes padded to
// 32) directly in per-lane WMMA B-fragment order, plus per-class constants,
// and copy mean/var to the output tail.
//
// ws layout (floats):
//   [0,      4096) : float2 bstream[2][32][32]  (nt, kk, lane) -> (W[kp], W[kp+1])
//   [4096,   4128) : const[32]  (c >= 21 -> 0)
// ---------------------------------------------------------------------------
__global__ void maha_prep_kernel(const float* __restrict__ mean,
                                 const float* __restrict__ var,
                                 float* __restrict__ ws,
                                 float* __restrict__ mean_out,
                                 float* __restrict__ var_out) {
  const int tid = threadIdx.x;  // 256 threads, 1 block
  for (int idx = tid; idx < 2048; idx += 256) {
    const int lane = idx & 31;
    const int kk   = (idx >> 5) & 31;
    const int nt   = idx >> 10;
    const int lh   = lane >> 4;
    const int l15  = lane & 15;
    const int c    = nt * 16 + l15;
    const int kp   = 4 * kk + 2 * lh;   // even augmented index held by this lane
    float we = 0.0f, wo = 0.0f;
    if (c < CC) {
      const int k   = kp >> 1;
      const float v = var[c * DD + k] + EPS_F;
      we = -0.5f / v;                  // weight for x^2 (even augmented row)
      wo = mean[c * DD + k] / v;       // weight for x   (odd augmented row)
    }
    ws[2 * idx]     = we;
    ws[2 * idx + 1] = wo;
  }
  if (tid < 32) {
    float s = 0.0f;
    if (tid < CC) {
      for (int k = 0; k < DD; ++k) {
        const float m = mean[tid * DD + k];
        const float v = var[tid * DD + k] + EPS_F;
        s += m * m / (2.0f * v) + 0.5f * logf(v);
      }
    }
    ws[4096 + tid] = s;
  }
  for (int i = tid; i < CC * DD; i += 256) {
    mean_out[i] = mean[i];
    var_out[i]  = var[i];
  }
}

// ---------------------------------------------------------------------------
// Main kernel: 128 threads (4 waves) per block; block = 128 consecutive w
// positions of one (n, h) row.
//   Phase 1: coalesced transpose-load of ft tile into padded LDS
//   Phase 2: coalesced streaming store of x  (exactly the transposed tile)
//   Phase 3: augmented-K f32 WMMA GEMM vs 21 classes (padded to 32)
// ---------------------------------------------------------------------------
__global__ __launch_bounds__(TILE) void maha_wmma_kernel(
    const float* __restrict__ ft, const float* __restrict__ ws,
    float* __restrict__ out0, float* __restrict__ x_out) {
  __shared__ float sX[TILE * SXS];

  const int t = threadIdx.x;
  const long long p_base = (long long)blockIdx.x * TILE;
  const int n  = (int)(p_base >> 16);
  const int hi = (int)((p_base >> 8) & 255);
  const int w0 = (int)(p_base & 255);  // 0 or 128

  // ---- Phase 1: load ft[n, k, hi, w0 + t] for k = 0..63 (coalesced in w)
  const float* ftp = ft + (long long)n * DD * HW + (long long)hi * WW + w0 + t;
#pragma unroll 8
  for (int k = 0; k < DD; ++k) {
    sX[t * SXS + k] = __builtin_nontemporal_load(ftp + (long long)k * HW);
  }
  __syncthreads();

  // ---- Phase 2: x output (row-major [point][k]) — fully coalesced
  {
    float* xo = x_out + p_base * DD;
#pragma unroll 8
    for (int j = 0; j < DD; ++j) {
      const int i = t + TILE * j;
      __builtin_nontemporal_store(sX[(i >> 6) * SXS + (i & 63)], xo + i);
    }
  }

  // ---- Phase 3: WMMA GEMM over augmented K' = 128
  const int lane = t & 31;
  const int wv   = t >> 5;
  const int l15  = lane & 15;
  const int lh   = lane >> 4;

  const float2* __restrict__ bs = (const float2*)ws;  // B fragment stream
  const float* cst = ws + 4096;
  const float c0 = -cst[l15];
  const float c1 = -cst[16 + l15];
  v8f acc00 = {c0, c0, c0, c0, c0, c0, c0, c0};  // rows [wv*32, +16), classes 0..15
  v8f acc01 = {c1, c1, c1, c1, c1, c1, c1, c1};  // rows [wv*32, +16), classes 16..31
  v8f acc10 = acc00;                             // rows [wv*32+16, +16), classes 0..15
  v8f acc11 = acc01;                             // rows [wv*32+16, +16), classes 16..31

  const int row0 = wv * 32 + l15;
#pragma unroll 4
  for (int kk = 0; kk < 32; ++kk) {
    const int k = 2 * kk + lh;                   // feature index for this lane half
    const float xv0 = sX[row0 * SXS + k];
    const float xv1 = sX[(row0 + 16) * SXS + k];
    const v2f a0 = {xv0 * xv0, xv0};             // augmented A fragment: (x^2, x)
    const v2f a1 = {xv1 * xv1, xv1};
    const float2 bf0 = bs[kk * 32 + lane];
    const float2 bf1 = bs[1024 + kk * 32 + lane];
    const v2f b0 = {bf0.x, bf0.y};
    const v2f b1 = {bf1.x, bf1.y};
    acc00 = wmma_f32x4(a0, b0, acc00);
    acc01 = wmma_f32x4(a0, b1, acc01);
    acc10 = wmma_f32x4(a1, b0, acc10);
    acc11 = wmma_f32x4(a1, b1, acc11);
  }

  // ---- Stores: out0[(n*21 + c)*HW + hi*WW + w]
  const long long obase = (long long)(n * CC) * HW + (long long)hi * WW + w0;
  const int ccol0 = l15;        // always < 21
  const int ccol1 = 16 + l15;   // valid only if < 21
#pragma unroll
  for (int r = 0; r < 8; ++r) {
    const int M   = 8 * lh + r;       // row within 16x16 D tile (C/D layout)
    const int wc0 = wv * 32 + M;      // point-tile 0
    const int wc1 = wc0 + 16;         // point-tile 1
    __builtin_nontemporal_store(acc00[r], out0 + obase + (long long)ccol0 * HW + wc0);
    __builtin_nontemporal_store(acc10[r], out0 + obase + (long long)ccol0 * HW + wc1);
    if (ccol1 < CC) {
      __builtin_nontemporal_store(acc01[r], out0 + obase + (long long)ccol1 * HW + wc0);
      __builtin_nontemporal_store(acc11[r], out0 + obase + (long long)ccol1 * HW + wc1);
    }
  }
}

extern "C" void kernel_launch(void* const* d_in, const int* in_sizes, int n_in,
                              void* d_out, int out_size, void* d_ws, size_t ws_size,
                              hipStream_t stream) {
  const float* ft   = (const float*)d_in[0];
  const float* mean = (const float*)d_in[1];
  const float* var  = (const float*)d_in[2];

  float* out      = (float*)d_out;
  float* out0     = out;                                   // [8, 21, 256, 256]
  float* x_out    = out + (long long)NB * CC * HW;         // [524288, 64]
  float* mean_out = x_out + (long long)NPTS * DD;          // [21, 64]
  float* var_out  = mean_out + CC * DD;                    // [21, 64]
  float* ws       = (float*)d_ws;

  maha_prep_kernel<<<1, 256, 0, stream>>>(mean, var, ws, mean_out, var_out);
  maha_wmma_kernel<<<NPTS / TILE, TILE, 0, stream>>>(ft, ws, out0, x_out);
}